// GraphConv_38611755991786
// MI455X (gfx1250) — compile-verified
//
#include <hip/hip_runtime.h>
#include <hip/hip_bf16.h>

// CDNA5 / gfx1250, wave32.
typedef __attribute__((ext_vector_type(2))) float v2f;
typedef __attribute__((ext_vector_type(8))) float v8f;

#define N_NODES 50000
#define N_EDGES 800000
#define D 128   // D_IN == D_OUT == 128

// -------------------------------------------------------------------------
// Kernel 1: h = x @ W using V_WMMA_F32_16X16X4_F32 (fp32 matrix pipe).
// One wave per 16x16 output tile; 8 waves per block cover the full 128-wide
// row stripe. K=128 consumed in steps of 4 -> 32 WMMAs per tile.
// -------------------------------------------------------------------------
__global__ __launch_bounds__(256) void gc_gemm_wmma(const float* __restrict__ x,
                                                    const float* __restrict__ w,
                                                    float* __restrict__ h) {
  const int lane = threadIdx.x & 31;
  const int wave = threadIdx.x >> 5;          // 0..7 -> N tile
  const int m0   = blockIdx.x * 16;           // row tile (50000/16 = 3125 exact)
  const int n0   = wave * 16;
  const int half = lane >> 4;                 // 0: lanes 0-15, 1: lanes 16-31
  const int l16  = lane & 15;
  const int arow = m0 + l16;                  // A-matrix row for this lane
  const int koff = half * 2;                  // upper half-lanes own K+2,K+3
  const int bcol = n0 + l16;                  // B/C/D column for this lane

  v8f acc = {};
#pragma unroll
  for (int k = 0; k < D; k += 4) {
    // A fragment (16x4): lane half selects K pair {k,k+1} or {k+2,k+3}
    const float* ap = x + arow * D + k + koff;
    v2f a;
    a.x = ap[0];
    a.y = ap[1];
    // B fragment (4x16): VGPR0 = row K (+2 for upper half), VGPR1 = next row
    const float* bp = w + (k + koff) * D + bcol;
    v2f b;
    b.x = bp[0];
    b.y = bp[D];
    acc = __builtin_amdgcn_wmma_f32_16x16x4_f32(
        /*neg_a=*/false, a, /*neg_b=*/false, b,
        /*c_mod=*/(short)0, acc, /*reuse_a=*/false, /*reuse_b=*/false);
  }

  // D-matrix layout: VGPR i -> M = m0 + i (+8 for upper half-lanes), N = bcol
#pragma unroll
  for (int i = 0; i < 8; ++i) {
    const int row = m0 + i + half * 8;
    h[row * D + bcol] = acc[i];
  }
}

// -------------------------------------------------------------------------
// Kernel 2: out[n][:] = bias[:]  (broadcast init; float4 stores)
// -------------------------------------------------------------------------
__global__ __launch_bounds__(256) void gc_init_bias(const float* __restrict__ bias,
                                                    float* __restrict__ out) {
  const int idx   = blockIdx.x * blockDim.x + threadIdx.x;   // float4 index
  const int total = N_NODES * (D / 4);
  if (idx < total) {
    const float4 b = reinterpret_cast<const float4*>(bias)[idx & (D / 4 - 1)];
    reinterpret_cast<float4*>(out)[idx] = b;
  }
}

// -------------------------------------------------------------------------
// Kernel 3: out[row[e]] += val[e] * h[col[e]]  for all edges.
// One wave32 per edge: lane owns 4 contiguous channels -> coalesced float4
// gather (512B/row, L2-resident) + 4 native global_atomic_add_f32 per lane
// (no-return form: fire-and-forget L2-side fp adds, tracked via STOREcnt).
// Edge lists are streamed once -> non-temporal loads keep L2 for h/out.
// -------------------------------------------------------------------------
__global__ __launch_bounds__(256) void gc_edge_scatter(const float* __restrict__ h,
                                                       const int* __restrict__ rows,
                                                       const int* __restrict__ cols,
                                                       const float* __restrict__ vals,
                                                       float* __restrict__ out) {
  const int lane   = threadIdx.x & 31;
  const int wave   = (blockIdx.x * blockDim.x + threadIdx.x) >> 5;
  const int nwaves = (gridDim.x * blockDim.x) >> 5;

  for (int e = wave; e < N_EDGES; e += nwaves) {
    const int   r = __builtin_nontemporal_load(rows + e);
    const int   c = __builtin_nontemporal_load(cols + e);
    const float v = __builtin_nontemporal_load(vals + e);

    const float4 hv = reinterpret_cast<const float4*>(h + (long)c * D)[lane];
    float* op = out + (long)r * D + lane * 4;

    // HW float atomics: GLOBAL_ATOMIC_ADD_F32 (device-scope, no CAS loop).
    unsafeAtomicAdd(op + 0, v * hv.x);
    unsafeAtomicAdd(op + 1, v * hv.y);
    unsafeAtomicAdd(op + 2, v * hv.z);
    unsafeAtomicAdd(op + 3, v * hv.w);
  }
}

// -------------------------------------------------------------------------
// Launcher. Inputs (setup_inputs order):
//   0: x [50000*128] f32, 1: weight [128*128] f32, 2: bias [128] f32,
//   3: edge_rows [800000] i32, 4: edge_cols [800000] i32, 5: edge_vals f32
// d_out: 50000*128 f32. d_ws: holds h (25.6 MB).
// -------------------------------------------------------------------------
extern "C" void kernel_launch(void* const* d_in, const int* in_sizes, int n_in,
                              void* d_out, int out_size, void* d_ws, size_t ws_size,
                              hipStream_t stream) {
  const float* x    = (const float*)d_in[0];
  const float* w    = (const float*)d_in[1];
  const float* bias = (const float*)d_in[2];
  const int*   er   = (const int*)d_in[3];
  const int*   ec   = (const int*)d_in[4];
  const float* ev   = (const float*)d_in[5];
  float*       out  = (float*)d_out;
  float*       h    = (float*)d_ws;   // N_NODES * D floats

  // 1) h = x @ W   (3125 row-tiles x 8 col-tiles; 8 waves/block)
  gc_gemm_wmma<<<N_NODES / 16, 256, 0, stream>>>(x, w, h);

  // 2) out = bias (broadcast)
  const int initv = N_NODES * (D / 4);
  gc_init_bias<<<(initv + 255) / 256, 256, 0, stream>>>(bias, out);

  // 3) out[row] += val * h[col]
  gc_edge_scatter<<<4096, 256, 0, stream>>>(h, er, ec, ev, out);
}